// ICRewardModel_26182120637033
// MI455X (gfx1250) — compile-verified
//
#include <hip/hip_runtime.h>

// ICRewardModel: out[b] = sum_s sigmoid(x[b,s,:]·W[0,:] + b0) * (x[b,s,:]·W[1,:] + b1) * (s < len[b])
//
// Memory-bound: 256 MiB of x streamed once -> ~11.5 us floor @ 23.3 TB/s.
// Dot products via V_WMMA_F32_16X16X4_F32: A = W tile (rows 0..1 live, rows >=2
// read a zeroed LDS row -> unconditional ds_load, no EXEC masking), B = x tile
// (4 d-values x 16 tokens). Two accumulators break the WMMA RAW chain.

typedef __attribute__((ext_vector_type(2))) float v2f;
typedef __attribute__((ext_vector_type(8))) float v8f;

#define BATCH   16
#define SEQ     4096
#define DIM     1024
#define NTOK    (BATCH * SEQ)           // 65536
#define WAVES_PER_BLOCK 8
#define TOK_PER_WAVE    16
#define TOK_PER_BLOCK   (WAVES_PER_BLOCK * TOK_PER_WAVE)   // 128
#define NBLOCKS         (NTOK / TOK_PER_BLOCK)             // 512

__global__ void ic_zero_out(float* __restrict__ out) {
    if (threadIdx.x < BATCH) out[threadIdx.x] = 0.0f;
}

__global__ __launch_bounds__(256) void ic_reward_kernel(
    const float* __restrict__ x,        // (B, S, D)
    const float* __restrict__ W,        // (2, D)
    const float* __restrict__ bias,     // (2,)
    const int*   __restrict__ lengths,  // (B,)
    float*       __restrict__ out)      // (B,)
{
    // Rows 0,1: W. Row 2: zeros (read by A-matrix lanes m >= 2).
    __shared__ float lds_w[3 * DIM];    // 12 KB
    for (int i = threadIdx.x; i < 3 * DIM; i += 256)
        lds_w[i] = (i < 2 * DIM) ? W[i] : 0.0f;
    __syncthreads();

    const int lane  = threadIdx.x & 31;
    const int wave  = threadIdx.x >> 5;
    const int m     = lane & 15;             // A lane: M row / B lane: N column
    const int khalf = (lane >> 4) << 1;      // lanes 0-15 -> K={0,1}; 16-31 -> K={2,3}

    const int  tok_base = (blockIdx.x * WAVES_PER_BLOCK + wave) * TOK_PER_WAVE;
    const int  tok      = tok_base + m;      // token this lane represents (column N)
    const float* xrow   = x + (long)tok * DIM + khalf;
    const float* wrow   = lds_w + (m < 2 ? m : 2) * DIM + khalf;   // always valid

    v8f c0 = {}, c1 = {};
    for (int d0 = 0; d0 < DIM; d0 += 32) {
        // Pull the stream ~1KB ahead (speculative; dropped silently past the end).
        __builtin_prefetch(xrow + d0 + 256, 0, 3);

        v2f bf[8], af[8];
        #pragma unroll
        for (int j = 0; j < 8; ++j) {
            bf[j] = *(const v2f*)(xrow + d0 + 4 * j);   // global_load_b64 (coalesces over k-walk)
            af[j] = *(const v2f*)(wrow + d0 + 4 * j);   // ds_load_b64, unconditional
        }
        #pragma unroll
        for (int j = 0; j < 8; j += 2) {
            // (neg_a, A, neg_b, B, c_mod, C, reuse_a, reuse_b)
            c0 = __builtin_amdgcn_wmma_f32_16x16x4_f32(
                     false, af[j],     false, bf[j],     (short)0, c0, false, false);
            c1 = __builtin_amdgcn_wmma_f32_16x16x4_f32(
                     false, af[j + 1], false, bf[j + 1], (short)0, c1, false, false);
        }
    }

    // C layout: VGPR0 lanes 0-15 = D[M=0][N=lane] (gate logit),
    //           VGPR1 lanes 0-15 = D[M=1][N=lane] (reward). Lanes 16-31 unused rows.
    const int   bidx = tok >> 12;            // token / 4096
    const int   sidx = tok & (SEQ - 1);
    const float g    = (c0[0] + c1[0]) + bias[0];
    const float r    = (c0[1] + c1[1]) + bias[1];
    const float gate = 1.0f / (1.0f + __expf(-g));
    float val = (sidx < lengths[bidx]) ? (gate * r) : 0.0f;
    if (lane >= 16) val = 0.0f;

    // Butterfly reduce across the wave (upper half already zero).
    #pragma unroll
    for (int off = 16; off > 0; off >>= 1)
        val += __shfl_xor(val, off, 32);

    if (lane == 0) atomicAdd(&out[bidx], val);
}

extern "C" void kernel_launch(void* const* d_in, const int* in_sizes, int n_in,
                              void* d_out, int out_size, void* d_ws, size_t ws_size,
                              hipStream_t stream) {
    const float* x       = (const float*)d_in[0];
    const float* W       = (const float*)d_in[1];
    const float* bias    = (const float*)d_in[2];
    const int*   lengths = (const int*)d_in[3];
    float*       out     = (float*)d_out;

    ic_zero_out<<<1, 32, 0, stream>>>(out);
    ic_reward_kernel<<<NBLOCKS, 256, 0, stream>>>(x, W, bias, lengths, out);
}